// EdgeMLP_30580167147630
// MI455X (gfx1250) — compile-verified
//
#include <hip/hip_runtime.h>
#include <hip/hip_bf16.h>

// EdgeMLP on MI455X (gfx1250), factored form:
//   A'[n] = emb[n] @ W1[0:32,:]  + b1      (cols   0..63 of AC)
//   C [n] = emb[n] @ W1[32:64,:]           (cols 64..127 of AC)
//   adj[b,r,c] = sigmoid( sum_h relu(A'[lo][h] + C[hi][h]) * W2[h] + b2 ),
//                lo=min(r,c), hi=max(r,c), 0 on diagonal.
// Stage 1: fp32 WMMA GEMM (V_WMMA_F32_16X16X4_F32), 8192x32 @ 32x128.
// Stage 2: LDS-tiled elementwise MLP tail, coalesced full-matrix stores.

typedef __attribute__((ext_vector_type(2))) float v2f;
typedef __attribute__((ext_vector_type(8))) float v8f;

#define EMBED   32
#define HIDDEN  64
#define NBATCH  16
#define NNODES  512
#define BN      (NBATCH * NNODES)   // 8192 rows
#define NCOLS   128                 // [A' | C] columns
#define NTILES_N (NCOLS / 16)       // 8
#define NTILES_M (BN / 16)          // 512

// ---------------------------------------------------------------------------
// Stage 1: AC[8192][128] = emb[8192][32] @ W1cat[32][128]  (+ b1 on A' half)
// W1cat[k][col] = (col < 64) ? W1[k][col] : W1[32+k][col-64]
// One wave per 16x16 output tile; K=32 covered by 8 x V_WMMA_F32_16X16X4_F32.
// ---------------------------------------------------------------------------
__global__ __launch_bounds__(256) void edge_mlp_stage1(
    const float* __restrict__ emb,   // BN x 32
    const float* __restrict__ W1,    // 64 x 64
    const float* __restrict__ b1,    // 64
    float* __restrict__ AC)          // BN x 128
{
    const int wave = threadIdx.x >> 5;
    const int lane = threadIdx.x & 31;
    const int tile = blockIdx.x * 8 + wave;     // 0..4095
    const int mtile = tile >> 3;                // 0..511
    const int ntile = tile & 7;                 // 0..7

    // A fragment addressing (ISA 7.12.2, 32-bit A 16x4):
    //   lanes 0-15: row m=lane,    K = {0,1} in VGPR0,1
    //   lanes 16-31: row m=lane-16, K = {2,3}
    const int m     = lane & 15;
    const int khalf = (lane >> 4) << 1;         // 0 or 2
    const int n     = lane & 15;
    const int col   = ntile * 16 + n;           // 0..127 of AC

    const float* __restrict__ arow = emb + (mtile * 16 + m) * EMBED;
    // B operand column pointer into W1 (stride 64 floats between k's)
    const int  wcol = (col < HIDDEN) ? col : (col - HIDDEN);
    const int  krow = (col < HIDDEN) ? 0 : 32;
    const float* __restrict__ wptr = W1 + krow * HIDDEN + wcol;

    v8f acc = {};
#pragma unroll
    for (int kk = 0; kk < 8; ++kk) {
        const int k0 = kk * 4 + khalf;
        v2f a, b;
        a.x = arow[k0];
        a.y = arow[k0 + 1];
        b.x = wptr[k0 * HIDDEN];
        b.y = wptr[(k0 + 1) * HIDDEN];
        acc = __builtin_amdgcn_wmma_f32_16x16x4_f32(
            /*neg_a=*/false, a, /*neg_b=*/false, b,
            /*c_mod=*/(short)0, acc, /*reuse_a=*/false, /*reuse_b=*/false);
    }

    const float bias = (col < HIDDEN) ? b1[col] : 0.0f;
    float* __restrict__ dst = AC + (mtile * 16) * NCOLS + col;
#pragma unroll
    for (int v = 0; v < 8; ++v) {
        const int row = v + ((lane >> 4) << 3);   // D layout: M = v (+8 for hi lanes)
        dst[row * NCOLS] = acc[v] + bias;
    }
}

// ---------------------------------------------------------------------------
// Stage 2: 32x32 output tile per block, both triangles computed directly.
// LDS holds 64 AC rows (32 r-rows, 32 c-rows), padded stride 132 floats
// (132 % 64 == 4 -> no 64-bank stride-64 conflicts on ds_load_b128).
// ---------------------------------------------------------------------------
#define TS   32
#define LROW 132

__global__ __launch_bounds__(256) void edge_mlp_stage2(
    const float* __restrict__ AC,    // BN x 128
    const float* __restrict__ W2,    // 64
    const float* __restrict__ b2,    // 1
    float* __restrict__ out)         // 16 x 512 x 512
{
    __shared__ __align__(16) float sAC[64 * LROW];
    __shared__ float sW2[HIDDEN];
    __shared__ float sB2;

    const int t     = threadIdx.x;
    const int b     = blockIdx.z;
    const int rbase = blockIdx.y * TS;
    const int cbase = blockIdx.x * TS;

    // Stage 64 rows x 128 floats into LDS as float4 (8 per thread).
    for (int idx = t; idx < 64 * 32; idx += 256) {
        const int lrow = idx >> 5;
        const int seg  = idx & 31;
        const int grow = (lrow < TS) ? (rbase + lrow) : (cbase + (lrow - TS));
        const float4 v =
            ((const float4*)(AC + (size_t)(b * NNODES + grow) * NCOLS))[seg];
        *(float4*)&sAC[lrow * LROW + seg * 4] = v;
    }
    if (t < HIDDEN) sW2[t] = W2[t];
    if (t == 0)     sB2 = b2[0];
    __syncthreads();

    const int tx = t & 31;      // column within tile (coalesced stores)
    const int ty = t >> 5;      // wave id -> row group
    const int cg = cbase + tx;
    const float b2v = sB2;

#pragma unroll
    for (int k = 0; k < 4; ++k) {
        const int rl = ty + 8 * k;            // local row 0..31
        const int rg = rbase + rl;
        float res = 0.0f;
        if (rg != cg) {
            int lo_loc, hi_loc;
            if (rg < cg) { lo_loc = rl;       hi_loc = TS + tx; }
            else         { lo_loc = TS + tx;  hi_loc = rl;      }
            const float* __restrict__ aRow = &sAC[lo_loc * LROW];       // A' cols 0..63
            const float* __restrict__ cRow = &sAC[hi_loc * LROW + 64];  // C  cols 0..63
            float acc = b2v;
#pragma unroll
            for (int h = 0; h < HIDDEN; h += 4) {
                const float4 av = *(const float4*)(aRow + h);
                const float4 cv = *(const float4*)(cRow + h);
                const float t0 = fmaxf(av.x + cv.x, 0.0f);
                const float t1 = fmaxf(av.y + cv.y, 0.0f);
                const float t2 = fmaxf(av.z + cv.z, 0.0f);
                const float t3 = fmaxf(av.w + cv.w, 0.0f);
                acc = fmaf(t0, sW2[h + 0], acc);
                acc = fmaf(t1, sW2[h + 1], acc);
                acc = fmaf(t2, sW2[h + 2], acc);
                acc = fmaf(t3, sW2[h + 3], acc);
            }
            res = 1.0f / (1.0f + __expf(-acc));
        }
        out[((size_t)b * NNODES + rg) * NNODES + cg] = res;
    }
}

// ---------------------------------------------------------------------------
extern "C" void kernel_launch(void* const* d_in, const int* in_sizes, int n_in,
                              void* d_out, int out_size, void* d_ws, size_t ws_size,
                              hipStream_t stream) {
    const float* emb = (const float*)d_in[0];  // 8192 x 32
    const float* W1  = (const float*)d_in[1];  // 64 x 64
    const float* b1  = (const float*)d_in[2];  // 64
    const float* W2  = (const float*)d_in[3];  // 64
    const float* b2  = (const float*)d_in[4];  // 1
    float* out = (float*)d_out;                // 16 x 512 x 512
    float* AC  = (float*)d_ws;                 // needs 8192*128*4 = 4 MB scratch

    // 4096 WMMA tiles, 8 waves (256 threads) per block -> 512 blocks.
    edge_mlp_stage1<<<dim3(NTILES_M * NTILES_N / 8), dim3(256), 0, stream>>>(
        emb, W1, b1, AC);
    // 16x16 tiles of 32x32 outputs per batch.
    edge_mlp_stage2<<<dim3(NNODES / TS, NNODES / TS, NBATCH), dim3(256), 0,
                      stream>>>(AC, W2, b2, out);
}